// GCN_13889924235582
// MI455X (gfx1250) — compile-verified
//
#include <hip/hip_runtime.h>
#include <hip/hip_bf16.h>

#define N_NODES 50000
#define N_EDGES 800000
#define NFEAT   256
#define NHID    64

typedef __attribute__((ext_vector_type(2))) float v2f;
typedef __attribute__((ext_vector_type(8))) float v8f;

// ---- degree accumulation (deg_src / deg_dst live in ws, pre-zeroed) ----
__global__ void deg_kernel(const int* __restrict__ src, const int* __restrict__ dst,
                           float* __restrict__ deg_src, float* __restrict__ deg_dst) {
    int e = blockIdx.x * blockDim.x + threadIdx.x;
    if (e < N_EDGES) {
        atomicAdd(&deg_src[src[e]], 1.0f);
        atomicAdd(&deg_dst[dst[e]], 1.0f);
    }
}

// deg -> rsqrt(max(deg,1)) in place (covers both arrays contiguously)
__global__ void norm_kernel(float* __restrict__ deg, int n) {
    int i = blockIdx.x * blockDim.x + threadIdx.x;
    if (i < n) deg[i] = rsqrtf(fmaxf(deg[i], 1.0f));
}

// ---- layer-1 GEMM via V_WMMA_F32_16X16X4_F32: h1 = (x * norm_src) @ W1 ----
// One wave computes a full 16x64 output tile (4 N-tiles, 64 K-steps of 4).
// 50000 rows = 3125 exact M-tiles; 4 waves per block.
__global__ void __launch_bounds__(128)
gemm1_wmma(const float* __restrict__ x, const float* __restrict__ W1,
           const float* __restrict__ norm_src, float* __restrict__ h1) {
    const int wave  = threadIdx.x >> 5;
    const int lane  = threadIdx.x & 31;
    const int tileM = blockIdx.x * 4 + wave;
    if (tileM * 16 >= N_NODES) return;          // whole-wave exit only

    const int row0 = tileM * 16;
    const int half = lane >> 4;                 // 0: K={k0,k0+1}, 1: K={k0+2,k0+3}
    const int l    = lane & 15;
    const int r    = row0 + l;                  // A-matrix row for this lane
    const float s  = norm_src[r];
    const float* xrow = x + (size_t)r * NFEAT + half * 2;

    v8f acc0 = {}; v8f acc1 = {}; v8f acc2 = {}; v8f acc3 = {};

#pragma unroll 2
    for (int k0 = 0; k0 < NFEAT; k0 += 4) {
        v2f a;
        a.x = s * xrow[k0];
        a.y = s * xrow[k0 + 1];
        const float* wk = W1 + (size_t)(k0 + half * 2) * NHID + l;
        v2f b0; b0.x = wk[0];  b0.y = wk[NHID + 0];
        v2f b1; b1.x = wk[16]; b1.y = wk[NHID + 16];
        v2f b2; b2.x = wk[32]; b2.y = wk[NHID + 32];
        v2f b3; b3.x = wk[48]; b3.y = wk[NHID + 48];
        acc0 = __builtin_amdgcn_wmma_f32_16x16x4_f32(false, a, false, b0, (short)0, acc0, false, false);
        acc1 = __builtin_amdgcn_wmma_f32_16x16x4_f32(false, a, false, b1, (short)0, acc1, false, false);
        acc2 = __builtin_amdgcn_wmma_f32_16x16x4_f32(false, a, false, b2, (short)0, acc2, false, false);
        acc3 = __builtin_amdgcn_wmma_f32_16x16x4_f32(false, a, false, b3, (short)0, acc3, false, false);
    }

    // D layout: VGPR i -> row row0 + half*8 + i, col t*16 + l
#pragma unroll
    for (int i = 0; i < 8; ++i) {
        int rr = row0 + half * 8 + i;
        float* o = h1 + (size_t)rr * NHID + l;
        o[0]  = acc0[i];
        o[16] = acc1[i];
        o[32] = acc2[i];
        o[48] = acc3[i];
    }
}

// ---- layer-1 edge gather + scatter-add: agg1[dst] += h1[src]  (16 lanes/edge, float4) ----
__global__ void scatter1(const float* __restrict__ h1, const int* __restrict__ src,
                         const int* __restrict__ dst, float* __restrict__ agg1) {
    long long t = (long long)blockIdx.x * blockDim.x + threadIdx.x;
    if (t >= (long long)N_EDGES * 16) return;
    int e = (int)(t >> 4);
    int f = (int)(t & 15) * 4;
    int s = src[e], d = dst[e];
    const float4 v = *(const float4*)(h1 + (size_t)s * NHID + f);
    float* o = agg1 + (size_t)d * NHID + f;
    atomicAdd(o + 0, v.x);
    atomicAdd(o + 1, v.y);
    atomicAdd(o + 2, v.z);
    atomicAdd(o + 3, v.w);
}

// ---- relu(agg1 * norm_dst + b1), in place ----
__global__ void finalize1(float* __restrict__ agg1, const float* __restrict__ norm_dst,
                          const float* __restrict__ b1) {
    int t = blockIdx.x * blockDim.x + threadIdx.x;
    if (t >= N_NODES * NHID) return;
    int n = t >> 6;
    int f = t & 63;
    float v = agg1[t] * norm_dst[n] + b1[f];
    agg1[t] = fmaxf(v, 0.0f);
}

// ---- layer-2 GEMM (64 -> 2), one thread per node ----
__global__ void gemm2(const float* __restrict__ out1, const float* __restrict__ W2,
                      const float* __restrict__ norm_src, float* __restrict__ h2) {
    int n = blockIdx.x * blockDim.x + threadIdx.x;
    if (n >= N_NODES) return;
    const float* row = out1 + (size_t)n * NHID;
    float a0 = 0.0f, a1 = 0.0f;
#pragma unroll
    for (int k = 0; k < NHID; ++k) {
        float v = row[k];
        a0 += v * W2[k * 2 + 0];
        a1 += v * W2[k * 2 + 1];
    }
    float s = norm_src[n];
    h2[n * 2 + 0] = s * a0;
    h2[n * 2 + 1] = s * a1;
}

// ---- layer-2 scatter-add ----
__global__ void scatter2(const float* __restrict__ h2, const int* __restrict__ src,
                         const int* __restrict__ dst, float* __restrict__ agg2) {
    int e = blockIdx.x * blockDim.x + threadIdx.x;
    if (e >= N_EDGES) return;
    int s = src[e], d = dst[e];
    atomicAdd(&agg2[d * 2 + 0], h2[s * 2 + 0]);
    atomicAdd(&agg2[d * 2 + 1], h2[s * 2 + 1]);
}

// ---- norm + bias + log_softmax over 2 classes ----
__global__ void finalize2(const float* __restrict__ agg2, const float* __restrict__ norm_dst,
                          const float* __restrict__ b2, float* __restrict__ out) {
    int n = blockIdx.x * blockDim.x + threadIdx.x;
    if (n >= N_NODES) return;
    float nd = norm_dst[n];
    float l0 = agg2[n * 2 + 0] * nd + b2[0];
    float l1 = agg2[n * 2 + 1] * nd + b2[1];
    float m  = fmaxf(l0, l1);
    float lse = m + logf(expf(l0 - m) + expf(l1 - m));
    out[n * 2 + 0] = l0 - lse;
    out[n * 2 + 1] = l1 - lse;
}

extern "C" void kernel_launch(void* const* d_in, const int* in_sizes, int n_in,
                              void* d_out, int out_size, void* d_ws, size_t ws_size,
                              hipStream_t stream) {
    (void)in_sizes; (void)n_in; (void)out_size; (void)ws_size;
    const float* x  = (const float*)d_in[0];
    const int*  src = (const int*)d_in[1];
    const int*  dst = (const int*)d_in[2];
    const float* W1 = (const float*)d_in[3];
    const float* b1 = (const float*)d_in[4];
    const float* W2 = (const float*)d_in[5];
    const float* b2 = (const float*)d_in[6];
    float* out = (float*)d_out;

    // Workspace layout (floats): total 6,700,000 floats = 26.8 MB
    float* ws       = (float*)d_ws;
    float* norm_src = ws;              // 50000  (deg_src then norm in place)
    float* norm_dst = ws + 50000;      // 50000  (deg_dst then norm in place)
    float* h1       = ws + 100000;     // 3,200,000
    float* agg1     = ws + 3300000;    // 3,200,000
    float* h2       = ws + 6500000;    // 100,000
    float* agg2     = ws + 6600000;    // 100,000

    // capture-safe zeroing of accumulation buffers (every call)
    hipMemsetAsync(norm_src, 0, 100000 * sizeof(float), stream);   // deg_src + deg_dst
    hipMemsetAsync(agg1,     0, 3200000 * sizeof(float), stream);
    hipMemsetAsync(agg2,     0, 100000 * sizeof(float), stream);

    deg_kernel<<<(N_EDGES + 255) / 256, 256, 0, stream>>>(src, dst, norm_src, norm_dst);
    norm_kernel<<<(100000 + 255) / 256, 256, 0, stream>>>(norm_src, 100000);
    gemm1_wmma<<<(3125 + 3) / 4, 128, 0, stream>>>(x, W1, norm_src, h1);
    scatter1<<<(N_EDGES * 16) / 256, 256, 0, stream>>>(h1, src, dst, agg1);
    finalize1<<<(N_NODES * NHID + 255) / 256, 256, 0, stream>>>(agg1, norm_dst, b1);
    gemm2<<<(N_NODES + 255) / 256, 256, 0, stream>>>(agg1, W2, norm_src, h2);
    scatter2<<<(N_EDGES + 255) / 256, 256, 0, stream>>>(h2, src, dst, agg2);
    finalize2<<<(N_NODES + 255) / 256, 256, 0, stream>>>(agg2, norm_dst, b2, out);
}